// KGEmbedding_35519379537955
// MI455X (gfx1250) — compile-verified
//
#include <hip/hip_runtime.h>

// Problem constants (match reference)
constexpr int B_ = 8;
constexpr int N_ = 1024;
constexpr int C_ = 256;
constexpr int R_ = 16;
constexpr int M_ = B_ * N_;   // 8192 rows for the fused (B*N, C) projections

typedef __attribute__((ext_vector_type(2))) float v2f;
typedef __attribute__((ext_vector_type(4))) float v4f;
typedef __attribute__((ext_vector_type(8))) float v8f;

// ---------------------------------------------------------------------------
// Projection: Y[m,n] = sum_c X[m,c] * W[n,c] + bias[n]   (NT GEMM, K = C = 256)
// One wave per 16x64 output strip: the A fragment (X rows, 1 load/step) is
// reused across four B fragments / four V_WMMA_F32_16X16X4_F32 accumulators.
//
// Fragment layouts (CDNA5 ISA 7.12.2, 32-bit, wave32):
//   A 16x4 : lanes 0-15 hold M=lane, K={k,k+1};  lanes 16-31 hold M=lane-16, K={k+2,k+3}
//   B 4x16 : lanes 0-15 hold N=lane, K={k,k+1};  lanes 16-31 hold N=lane-16, K={k+2,k+3}
//   C/D    : VGPR v -> M = v + 8*(lane>>4), N = lane&15
// ---------------------------------------------------------------------------
__global__ __launch_bounds__(32)
void proj_kernel(const float* __restrict__ X, const float* __restrict__ W,
                 const float* __restrict__ bias, float* __restrict__ Y) {
    const int m0   = blockIdx.x * 16;
    const int n0   = blockIdx.y * 64;
    const int lane = threadIdx.x;
    const int row   = lane & 15;
    const int khalf = lane >> 4;

    const float* xrow = X + (size_t)(m0 + row) * C_;  // A: X[m0+row][*]
    const float* w0   = W + (size_t)(n0 + row) * C_;  // B: W[n][k] rows

    v8f acc[4] = {{}, {}, {}, {}};
    #pragma unroll 4
    for (int k = 0; k < C_; k += 4) {
        const int kk = k + 2 * khalf;
        v2f a = *(const v2f*)(xrow + kk);
        #pragma unroll
        for (int s = 0; s < 4; ++s) {
            v2f b = *(const v2f*)(w0 + (size_t)(16 * s) * C_ + kk);
            acc[s] = __builtin_amdgcn_wmma_f32_16x16x4_f32(
                false, a, false, b, (short)0, acc[s], false, false);
        }
    }

    #pragma unroll
    for (int s = 0; s < 4; ++s) {
        const float bv = bias[n0 + 16 * s + row];  // bias depends on N = lane&15
        #pragma unroll
        for (int v = 0; v < 8; ++v) {
            const int i = v + 8 * khalf;
            Y[(size_t)(m0 + i) * C_ + n0 + 16 * s + row] = acc[s][v] + bv;
        }
    }
}

// ---------------------------------------------------------------------------
// Fused scores + relation-mask broadcast.
// Block = 256 threads = 8 waves; block owns a 16(i) x 32(j) tile for ALL
// batches and relations. Wave w == batch w computes two 16x16 score tiles
// (shared A loads) via fp32 WMMA; qh/kh reuse is L2-resident (8 MB each).
// R_map tile (16x32x16 = 32 KB) is issued into registers BEFORE the WMMA
// loop (latency hidden behind the matmul), then scattered conflict-free into
// LDS transposed to [r][i*32+j]. Epilogue: linear ds_load_b128 + 537 MB of
// full-128B-line-coalesced global_store_b128.
// ---------------------------------------------------------------------------
__global__ __launch_bounds__(256)
void scores_mask_kernel(const float* __restrict__ qh, const float* __restrict__ kh,
                        const float* __restrict__ Rmap, float* __restrict__ out) {
    __shared__ float rel_s[R_ * 512];  // [r][i*32+j]   32 KB
    __shared__ float sc_s[B_ * 512];   // [b][i*32+j]   16 KB

    const int j0 = blockIdx.x * 32;
    const int i0 = blockIdx.y * 16;
    const int t  = threadIdx.x;

    // ---- A: issue R_map tile loads into registers (pipelined ahead of WMMA)
    // 512 (i,j) cells, 2 per thread; 4x global_load_b128 each, coalesced.
    v4f rv[2][4];
    int cell[2];
    #pragma unroll
    for (int c = 0; c < 2; ++c) {
        cell[c] = t + 256 * c;
        const int ri = cell[c] >> 5, rj = cell[c] & 31;
        const float* rp = Rmap + ((size_t)(i0 + ri) * N_ + (j0 + rj)) * R_;
        #pragma unroll
        for (int rq = 0; rq < 4; ++rq)
            rv[c][rq] = *(const v4f*)(rp + 4 * rq);
    }

    // ---- B: WMMA scores, wave id == batch; two j-subtiles share A loads ----
    {
        const int b     = t >> 5;       // batch 0..7
        const int lane  = t & 31;
        const int row   = lane & 15;
        const int khalf = lane >> 4;

        const float* qrow  = qh + (size_t)(b * N_ + i0 + row) * C_;
        const float* krow0 = kh + (size_t)(b * N_ + j0 + row) * C_;
        const float* krow1 = krow0 + (size_t)16 * C_;

        v8f acc0 = {}, acc1 = {};
        #pragma unroll 4
        for (int k = 0; k < C_; k += 4) {
            const int kk = k + 2 * khalf;
            v2f a  = *(const v2f*)(qrow  + kk);
            v2f b0 = *(const v2f*)(krow0 + kk);
            v2f b1 = *(const v2f*)(krow1 + kk);
            acc0 = __builtin_amdgcn_wmma_f32_16x16x4_f32(
                false, a, false, b0, (short)0, acc0, false, false);
            acc1 = __builtin_amdgcn_wmma_f32_16x16x4_f32(
                false, a, false, b1, (short)0, acc1, false, false);
        }
        #pragma unroll
        for (int v = 0; v < 8; ++v) {
            const int i = v + 8 * khalf;
            sc_s[b * 512 + i * 32 + row]      = acc0[v] * 0.0625f;  // 1/sqrt(C)
            sc_s[b * 512 + i * 32 + 16 + row] = acc1[v] * 0.0625f;
        }
    }

    // ---- C: scatter rel registers to LDS, transposed r-major ----
    // Bank = cell mod 64: lanes hold consecutive cells -> conflict-free.
    #pragma unroll
    for (int c = 0; c < 2; ++c)
        #pragma unroll
        for (int rq = 0; rq < 4; ++rq)
            #pragma unroll
            for (int e = 0; e < 4; ++e)
                rel_s[(4 * rq + e) * 512 + cell[c]] = rv[c][rq][e];

    __syncthreads();

    // ---- D: broadcast multiply + full-line coalesced b128 store stream ----
    // Wave lanes cover i0..i0+3 x 8 j-quads => 4 complete 128B lines/store.
    {
        const int b2  = t >> 7;          // 0..1
        const int i   = (t >> 3) & 15;   // 0..15
        const int jq  = t & 7;           // j-quad 0..7
        const int idx = i * 32 + 4 * jq; // linear LDS index, 16B aligned

        #pragma unroll
        for (int bb = 0; bb < 4; ++bb) {
            const int b = b2 + 2 * bb;
            const v4f s = *(const v4f*)(&sc_s[b * 512 + idx]);
            float* obase = out
                + (size_t)(b * R_) * N_ * N_
                + (size_t)(i0 + i) * N_ + (j0 + 4 * jq);
            #pragma unroll
            for (int r = 0; r < R_; ++r) {
                const v4f m = *(const v4f*)(&rel_s[r * 512 + idx]);
                *(v4f*)(obase + (size_t)r * N_ * N_) = s * m;
            }
        }
    }
}

extern "C" void kernel_launch(void* const* d_in, const int* in_sizes, int n_in,
                              void* d_out, int out_size, void* d_ws, size_t ws_size,
                              hipStream_t stream) {
    (void)in_sizes; (void)n_in; (void)out_size; (void)ws_size;

    const float* q    = (const float*)d_in[0];  // (B,N,C)
    const float* k    = (const float*)d_in[1];  // (B,N,C)
    const float* Wq   = (const float*)d_in[2];  // (C,C)
    const float* bq   = (const float*)d_in[3];  // (C,)
    const float* Wk   = (const float*)d_in[4];  // (C,C)
    const float* bk   = (const float*)d_in[5];  // (C,)
    const float* Rmap = (const float*)d_in[6];  // (N,N,R)
    float* out = (float*)d_out;                 // (B,R,N,N)

    float* qh = (float*)d_ws;                   // 8 MB
    float* kh = qh + (size_t)M_ * C_;           // 8 MB

    dim3 gp(M_ / 16, C_ / 64, 1);
    proj_kernel<<<gp, 32, 0, stream>>>(q, Wq, bq, qh);
    proj_kernel<<<gp, 32, 0, stream>>>(k, Wk, bk, kh);

    dim3 gs(N_ / 32, N_ / 16, 1);
    scores_mask_kernel<<<gs, 256, 0, stream>>>(qh, kh, Rmap, out);
}